// MultiBoxLoss_90477781057959
// MI455X (gfx1250) — compile-verified
//
#include <hip/hip_runtime.h>
#include <hip/hip_bf16.h>
#include <stdint.h>

// ---------------------------------------------------------------------------
// MultiBox SSD loss for MI455X (gfx1250, wave32).
// Bandwidth-bound: dominant cost is one 181MB pass over conf_data (~8us @23.3TB/s).
// Hard-negative mining done via bit-pattern bisection (top-K SUM, no argsort).
// CDNA5 paths: v_wmma_f32_16x16x4_f32 reduction (A=ones trick, exact),
// global_load_async_to_lds_b128/b32 staging + s_wait_asynccnt in k_topk,
// global_prefetch_b8 in k_ce.
// ---------------------------------------------------------------------------

typedef float v2f __attribute__((ext_vector_type(2)));
typedef float v8f __attribute__((ext_vector_type(8)));

#define WAVE 32

__device__ __forceinline__ float waveAddF(float v) {
    for (int off = 16; off > 0; off >>= 1) v += __shfl_xor(v, off, WAVE);
    return v;
}
__device__ __forceinline__ int waveAddI(int v) {
    for (int off = 16; off > 0; off >>= 1) v += __shfl_xor(v, off, WAVE);
    return v;
}
__device__ __forceinline__ float waveMaxF(float v) {
    for (int off = 16; off > 0; off >>= 1) v = fmaxf(v, __shfl_xor(v, off, WAVE));
    return v;
}

#if defined(__gfx1250__) && \
    __has_builtin(__builtin_amdgcn_global_load_async_to_lds_b32) && \
    __has_builtin(__builtin_amdgcn_global_load_async_to_lds_b128)
#define HAVE_ASYNC_LDS 1
// Exact parameter types per hipcc diagnostic: global side is
// addrspace(1) (printed as __device__) pointer to vector_size(16) int / int.
typedef int vi4 __attribute__((vector_size(4 * sizeof(int))));
typedef __attribute__((address_space(1))) vi4* gvi4_t;
typedef __attribute__((address_space(3))) vi4* lvi4_t;
typedef __attribute__((address_space(1))) int* gi32_t;
typedef __attribute__((address_space(3))) int* li32_t;

__device__ __forceinline__ void async_copy_b128(const void* g, const void* l) {
    // Flat LDS addresses truncate to the LDS byte offset (ISA §10.2 aperture);
    // as3 pointers are 32-bit, so round-trip through unsigned.
    __builtin_amdgcn_global_load_async_to_lds_b128(
        (gvi4_t)(uintptr_t)g, (lvi4_t)(unsigned)(uintptr_t)l, 0, 0);
}
__device__ __forceinline__ void async_copy_b32(const void* g, const void* l) {
    __builtin_amdgcn_global_load_async_to_lds_b32(
        (gi32_t)(uintptr_t)g, (li32_t)(unsigned)(uintptr_t)l, 0, 0);
}
__device__ __forceinline__ void wait_async0() {
#if __has_builtin(__builtin_amdgcn_s_wait_asynccnt)
    __builtin_amdgcn_s_wait_asynccnt(0);
#else
    asm volatile("s_wait_asynccnt 0x0" ::: "memory");
#endif
}
#else
#define HAVE_ASYNC_LDS 0
#endif

// acc layout: [0]=loss_l_sum  [1]=ce_pos_sum  [2]=ce_topk_neg_sum  [3]=unused
__global__ void k_init(int* num_pos, float* acc, int B) {
    int i = blockIdx.x * blockDim.x + threadIdx.x;
    if (i < B) num_pos[i] = 0;
    if (i < 4) acc[i] = 0.0f;
}

// One workgroup per (b,t): argmax_p IoU(truth_bt, point_form(prior_p)).
// Tie rule: first (lowest) prior index, matching jnp.argmax(axis=1).
__global__ void k_best_prior(const float* __restrict__ targets,
                             const float* __restrict__ priors,
                             int* __restrict__ bpi, int P, int NO) {
    int b = blockIdx.x / NO, t = blockIdx.x % NO;
    const float* tg = targets + ((size_t)b * NO + t) * 5;
    float tx0 = tg[0], ty0 = tg[1], tx1 = tg[2], ty1 = tg[3];
    float area_t = (tx1 - tx0) * (ty1 - ty0);

    float best = -1.0f;
    int bestIdx = 0;
    for (int p = threadIdx.x; p < P; p += blockDim.x) {
        float cx = priors[p * 4 + 0], cy = priors[p * 4 + 1];
        float w  = priors[p * 4 + 2], h  = priors[p * 4 + 3];
        float px0 = cx - 0.5f * w, py0 = cy - 0.5f * h;
        float px1 = cx + 0.5f * w, py1 = cy + 0.5f * h;
        float ix = fmaxf(fminf(tx1, px1) - fmaxf(tx0, px0), 0.0f);
        float iy = fmaxf(fminf(ty1, py1) - fmaxf(ty0, py0), 0.0f);
        float inter = ix * iy;
        float iou = inter / (area_t + w * h - inter);
        if (iou > best) { best = iou; bestIdx = p; }   // strided asc => first max kept
    }
    __shared__ float bv[256];
    __shared__ int   bi[256];
    bv[threadIdx.x] = best;
    bi[threadIdx.x] = bestIdx;
    __syncthreads();
    for (int s = 128; s > 0; s >>= 1) {
        if ((int)threadIdx.x < s) {
            float ov = bv[threadIdx.x + s];
            int   oi = bi[threadIdx.x + s];
            if (ov > bv[threadIdx.x] ||
                (ov == bv[threadIdx.x] && oi < bi[threadIdx.x])) {
                bv[threadIdx.x] = ov;
                bi[threadIdx.x] = oi;
            }
        }
        __syncthreads();
    }
    if (threadIdx.x == 0) bpi[blockIdx.x] = bi[0];
}

// Per (b,p): best truth + override, conf_t, encode, smooth-L1 for positives.
__global__ void k_match(const float* __restrict__ loc_data,
                        const float* __restrict__ targets,
                        const float* __restrict__ priors,
                        const int* __restrict__ bpi,
                        int* __restrict__ conf_t,
                        int* __restrict__ num_pos,
                        float* __restrict__ acc,
                        int P, int NO) {
    int b = blockIdx.y;
    int p = blockIdx.x * blockDim.x + threadIdx.x;

    __shared__ float sT[64 * 4];
    __shared__ float sL[64];
    __shared__ int   sB[64];
    for (int t = threadIdx.x; t < NO; t += blockDim.x) {
        const float* tg = targets + ((size_t)b * NO + t) * 5;
        sT[t * 4 + 0] = tg[0]; sT[t * 4 + 1] = tg[1];
        sT[t * 4 + 2] = tg[2]; sT[t * 4 + 3] = tg[3];
        sL[t] = tg[4];
        sB[t] = bpi[b * NO + t];
    }
    __syncthreads();

    float sl1 = 0.0f;
    int isPos = 0;
    if (p < P) {
        float cx = priors[p * 4 + 0], cy = priors[p * 4 + 1];
        float w  = priors[p * 4 + 2], h  = priors[p * 4 + 3];
        float px0 = cx - 0.5f * w, py0 = cy - 0.5f * h;
        float px1 = cx + 0.5f * w, py1 = cy + 0.5f * h;
        float area_p = w * h;

        float bestOv = -1.0f; int bestT = 0;
        for (int t = 0; t < NO; ++t) {
            float tx0 = sT[t * 4 + 0], ty0 = sT[t * 4 + 1];
            float tx1 = sT[t * 4 + 2], ty1 = sT[t * 4 + 3];
            float ix = fmaxf(fminf(tx1, px1) - fmaxf(tx0, px0), 0.0f);
            float iy = fmaxf(fminf(ty1, py1) - fmaxf(ty0, py0), 0.0f);
            float inter = ix * iy;
            float area_t = (tx1 - tx0) * (ty1 - ty0);
            float iou = inter / (area_t + area_p - inter);
            if (iou > bestOv) { bestOv = iou; bestT = t; }  // first max (argmax axis=0)
        }
        // scatter override: best prior of each truth forced to that truth, ov=2.0
        for (int t = 0; t < NO; ++t) {
            if (sB[t] == p) { bestT = t; bestOv = 2.0f; }   // last writer wins
        }
        int conf = (int)sL[bestT] + 1;
        if (bestOv < 0.5f) conf = 0;
        conf_t[(size_t)b * P + p] = conf;

        if (conf > 0) {
            isPos = 1;
            float m0 = sT[bestT * 4 + 0], m1 = sT[bestT * 4 + 1];
            float m2 = sT[bestT * 4 + 2], m3 = sT[bestT * 4 + 3];
            float g0 = ((m0 + m2) * 0.5f - cx) / (0.1f * w);
            float g1 = ((m1 + m3) * 0.5f - cy) / (0.1f * h);
            float g2 = __logf((m2 - m0) / w) / 0.2f;
            float g3 = __logf((m3 - m1) / h) / 0.2f;
            const float* ld = loc_data + ((size_t)b * P + p) * 4;
            float g[4] = {g0, g1, g2, g3};
            for (int k = 0; k < 4; ++k) {
                float d = ld[k] - g[k];
                float ad = fabsf(d);
                sl1 += (ad < 1.0f) ? 0.5f * d * d : ad - 0.5f;
            }
        }
    }
    // block reduce sl1 + pos count, one atomic each per block
    int lane = threadIdx.x & (WAVE - 1), wv = threadIdx.x >> 5;
    float ws = waveAddF(sl1);
    int   wc = waveAddI(isPos);
    __shared__ float rs[8];
    __shared__ int   rc[8];
    if (lane == 0) { rs[wv] = ws; rc[wv] = wc; }
    __syncthreads();
    if (threadIdx.x == 0) {
        float ts = 0.0f; int tc = 0;
        for (int i = 0; i < 8; ++i) { ts += rs[i]; tc += rc[i]; }
        if (ts != 0.0f) atomicAdd(&acc[0], ts);
        if (tc != 0)    atomicAdd(&num_pos[b], tc);
    }
}

// Bandwidth kernel: one wave32 per (b,p) row of C<=96 logits.
// ce = logsumexp(row) - row[conf_t]; lc = pos ? 0 : ce; accumulate pos-ce sum.
__global__ void k_ce(const float* __restrict__ conf,
                     const int* __restrict__ conf_t,
                     float* __restrict__ lc,
                     float* __restrict__ acc,
                     int rows, int C) {
    int lane = threadIdx.x & (WAVE - 1), wv = threadIdx.x >> 5;
    int row = blockIdx.x * 8 + wv;
    __shared__ float wpos[8];
    float myPos = 0.0f;
    if (row < rows) {
        const float* rp = conf + (size_t)row * C;
        __builtin_prefetch(rp + 8 * (size_t)C, 0, 1);   // next row group -> global_prefetch_b8
        const float NEG = -3.4e38f;
        float a0 = (lane      < C) ? rp[lane]      : NEG;
        float a1 = (lane + 32 < C) ? rp[lane + 32] : NEG;
        float a2 = (lane + 64 < C) ? rp[lane + 64] : NEG;
        float m = waveMaxF(fmaxf(a0, fmaxf(a1, a2)));
        float s = __expf(a0 - m) + __expf(a1 - m) + __expf(a2 - m);
        s = waveAddF(s);
        float lse = m + __logf(s);
        int ct = conf_t[row];
        float g = rp[ct];
        float ce = lse - g;
        bool pos = ct > 0;
        if (lane == 0) {
            lc[row] = pos ? 0.0f : ce;
            myPos = pos ? ce : 0.0f;
        }
    }
    if (lane == 0) wpos[wv] = myPos;
    __syncthreads();
    if (threadIdx.x == 0) {
        float t = 0.0f;
        for (int i = 0; i < 8; ++i) t += wpos[i];
        if (t != 0.0f) atomicAdd(&acc[1], t);
    }
}

// Per-batch: sum of top-K lc values via 31-step bisection on float bit pattern
// (lc >= 0, so uint order == float order). lc staged into LDS (P*4 ~ 35KB)
// via CDNA5 async global->LDS DMA when available (ASYNCcnt path).
__global__ void k_topk(const float* __restrict__ lc,
                       const int* __restrict__ num_pos,
                       float* __restrict__ acc, int P) {
    extern __shared__ float slc[];
    int tid = threadIdx.x, lane = tid & (WAVE - 1);
    const float* src = lc + (size_t)blockIdx.x * P;

#if HAVE_ASYNC_LDS
    if ((P & 3) == 0) {
        int nvec = P >> 2;                     // 16B chunks, row base 16B-aligned
        for (int i = tid; i < nvec; i += blockDim.x) {
            async_copy_b128(src + 4 * i, slc + 4 * i);
        }
    } else {
        for (int i = tid; i < P; i += blockDim.x) {
            async_copy_b32(src + i, slc + i);
        }
    }
    wait_async0();                             // s_wait_asynccnt 0
    __syncthreads();
#else
    for (int i = tid; i < P; i += blockDim.x) slc[i] = src[i];
    __syncthreads();
#endif

    int K = 3 * num_pos[blockIdx.x];
    if (K > P - 1) K = P - 1;
    if (K <= 0) return;                    // uniform across block

    __shared__ int cnt_s;
    unsigned lo = 0u, hi = 0x7F800000u;    // [0, +inf)
    while (lo < hi) {
        unsigned mid = lo + ((hi - lo) >> 1);
        float t = __uint_as_float(mid);
        if (tid == 0) cnt_s = 0;
        __syncthreads();
        int c = 0;
        for (int i = tid; i < P; i += blockDim.x) c += (slc[i] > t) ? 1 : 0;
        c = waveAddI(c);
        if (lane == 0) atomicAdd(&cnt_s, c);
        __syncthreads();
        int cnt = cnt_s;                   // uniform
        if (cnt < K) hi = mid; else lo = mid + 1;
        __syncthreads();                   // protect cnt_s from next-iter reset
    }
    float v = __uint_as_float(lo);         // K-th largest value

    float s = 0.0f; int c = 0;
    for (int i = tid; i < P; i += blockDim.x) {
        float x = slc[i];
        if (x > v) { s += x; c++; }
    }
    s = waveAddF(s); c = waveAddI(c);
    __shared__ float fs; __shared__ int cs;
    if (tid == 0) { fs = 0.0f; cs = 0; }
    __syncthreads();
    if (lane == 0) { atomicAdd(&fs, s); atomicAdd(&cs, c); }
    __syncthreads();
    if (tid == 0) atomicAdd(&acc[2], fs + (float)(K - cs) * v);
}

// Final: N = sum(num_pos) reduced with V_WMMA_F32_16X16X4_F32.
// A = ones(16x4) => D[m][n] = C[m][n] + sum_k B[k][n]; every row of D holds the
// 16 column sums, so a full 32-lane shuffle-sum of d[0] yields exactly 2*N in
// f32 (ints < 2^24, all adds exact). Layout-agnostic: any bijective placement
// of the 64 inputs into B sums correctly. Launched as a single wave32.
// Full-64 chunks use unconditional loads (clean codegen); masked tail only if
// B is not a multiple of 64.
__global__ void k_final(const int* __restrict__ num_pos,
                        const float* __restrict__ acc,
                        float* __restrict__ out, int B) {
    int lane = threadIdx.x;                // blockDim.x == 32, EXEC all ones
    v2f a; a[0] = 1.0f; a[1] = 1.0f;
    v8f c = {};
    int nfull = B & ~63;
    for (int base = 0; base < nfull; base += 64) {
        v2f bm;
        bm[0] = (float)num_pos[base + lane];
        bm[1] = (float)num_pos[base + lane + 32];
        c = __builtin_amdgcn_wmma_f32_16x16x4_f32(
                false, a, false, bm, (short)0, c, false, false);
    }
    if (nfull < B) {                       // uniform branch, EXEC stays full
        int i0 = nfull + lane, i1 = nfull + lane + 32;
        v2f bm;
        bm[0] = (i0 < B) ? (float)num_pos[i0] : 0.0f;
        bm[1] = (i1 < B) ? (float)num_pos[i1] : 0.0f;
        c = __builtin_amdgcn_wmma_f32_16x16x4_f32(
                false, a, false, bm, (short)0, c, false, false);
    }
    float p = c[0];
    for (int off = 16; off > 0; off >>= 1) p += __shfl_xor(p, off, WAVE);
    float N = p * 0.5f;                    // exact: halving a doubled exact sum
    if (lane == 0) {
        out[0] = acc[0] / N;               // loss_l / N
        out[1] = (acc[1] + acc[2]) / N;    // loss_c / N
    }
}

extern "C" void kernel_launch(void* const* d_in, const int* in_sizes, int n_in,
                              void* d_out, int out_size, void* d_ws, size_t ws_size,
                              hipStream_t stream) {
    const float* loc_data = (const float*)d_in[0];
    const float* conf     = (const float*)d_in[1];
    const float* priors   = (const float*)d_in[2];
    const float* targets  = (const float*)d_in[3];

    int P  = in_sizes[2] / 4;              // priors: (P,4)
    int BP = in_sizes[0] / 4;              // loc_data: (B,P,4)
    int B  = BP / P;
    int C  = in_sizes[1] / BP;             // conf_data: (B,P,C)
    int NO = in_sizes[3] / (5 * B);        // targets: (B,NO,5)

    // workspace layout
    float* lc      = (float*)d_ws;                 // B*P floats
    int*   conf_t  = (int*)(lc + BP);              // B*P ints
    int*   bpi     = conf_t + BP;                  // B*NO ints
    int*   num_pos = bpi + (size_t)B * NO;         // B ints
    float* acc     = (float*)(num_pos + B);        // 4 floats

    k_init<<<(B + 255) / 256, 256, 0, stream>>>(num_pos, acc, B);
    k_best_prior<<<B * NO, 256, 0, stream>>>(targets, priors, bpi, P, NO);
    dim3 gm((P + 255) / 256, B);
    k_match<<<gm, 256, 0, stream>>>(loc_data, targets, priors, bpi,
                                    conf_t, num_pos, acc, P, NO);
    k_ce<<<(BP + 7) / 8, 256, 0, stream>>>(conf, conf_t, lc, acc, BP, C);
    k_topk<<<B, 256, (size_t)P * sizeof(float), stream>>>(lc, num_pos, acc, P);
    k_final<<<1, 32, 0, stream>>>(num_pos, acc, (float*)d_out, B);
}